// AsymmetricWeightsDequantizer_29274497089862
// MI455X (gfx1250) — compile-verified
//
#include <hip/hip_runtime.h>

// ---------------------------------------------------------------------------
// AsymmetricWeightsDequantizer for MI455X (gfx1250)
//   out[o,i] = (float(w[o,i]) - zp[o]) * scale[o]
// Pure streaming op (~361 MB @ 23.3 TB/s => ~15.5 us floor). Memory-bound:
// use the CDNA5 async global->LDS path (ASYNCcnt) so in-flight data lives in
// the 320KB LDS instead of VGPRs, b128 transactions, NT cache policy on both
// the one-shot weight reads and the write-once output stores.
// ---------------------------------------------------------------------------

#define OUTF 4096
#define INF  11008
#define TPB  256
#define VEC  4
#define COLS_PER_IT (TPB * VEC)                         // 1024 columns / iter
#define NITER ((INF + COLS_PER_IT - 1) / COLS_PER_IT)   // 11
#define NS        4                                     // LDS stages
#define LOOKAHEAD 3                                     // async loads in flight

typedef int   __attribute__((ext_vector_type(4))) i32x4;
typedef float __attribute__((ext_vector_type(4))) f32x4;
typedef __attribute__((address_space(3))) i32x4 lds_i32x4;

template <int N>
__device__ __forceinline__ void wait_asynccnt_le() {
    asm volatile("s_wait_asynccnt %0" :: "n"(N) : "memory");
}

// global_load_async_to_lds_b128 vdst(lds byte addr), vaddr(u32 offset), saddr
// NT hint: weights are streamed exactly once -> don't rinse the 192MB L2.
__device__ __forceinline__ void async_load_b128_nt(unsigned lds_byte_addr,
                                                   unsigned global_byte_off,
                                                   const int* base) {
    asm volatile("global_load_async_to_lds_b128 %0, %1, %2 th:TH_LOAD_NT"
                 :
                 : "v"(lds_byte_addr), "v"(global_byte_off), "s"(base)
                 : "memory");
}

__global__ __launch_bounds__(TPB) void
AsymmetricWeightsDequantizer_29274497089862_kernel(const int* __restrict__ w,
                                                   const float* __restrict__ scale,
                                                   const float* __restrict__ zp,
                                                   float* __restrict__ out) {
    __shared__ i32x4 buf[NS][TPB];

    const int row = blockIdx.y;          // one block per output row
    const int tid = threadIdx.x;

    const float s = scale[row];          // wave-uniform -> scalar path
    const float z = zp[row];

    const unsigned rowbase = (unsigned)row * (unsigned)INF;  // element offset

    // Raw LDS byte offsets for each stage slot of this lane.
    unsigned lds_addr[NS];
#pragma unroll
    for (int p = 0; p < NS; ++p)
        lds_addr[p] = (unsigned)(size_t)(lds_i32x4*)&buf[p][tid];

    // --- prologue: fill the pipeline ------------------------------------
#pragma unroll
    for (int p = 0; p < LOOKAHEAD; ++p) {
        int col = p * COLS_PER_IT + tid * VEC;
        int sc  = (col > (INF - VEC)) ? (INF - VEC) : col;  // clamp, 16B aligned
        async_load_b128_nt(lds_addr[p], (rowbase + (unsigned)sc) * 4u, w);
    }

    // --- steady state ----------------------------------------------------
#pragma unroll
    for (int it = 0; it < NITER; ++it) {
        // Issue the load LOOKAHEAD iterations ahead (slot not in use).
        const int nxt = it + LOOKAHEAD;
        if (nxt < NITER) {
            int col = nxt * COLS_PER_IT + tid * VEC;
            int sc  = (col > (INF - VEC)) ? (INF - VEC) : col;
            async_load_b128_nt(lds_addr[nxt & (NS - 1)],
                               (rowbase + (unsigned)sc) * 4u, w);
        }

        // Async loads complete in order: load `it` done once outstanding
        // drops to min(LOOKAHEAD, NITER-1-it). `it` is uniform -> no diverge.
        const int rem = NITER - 1 - it;
        if (rem >= LOOKAHEAD)  wait_asynccnt_le<LOOKAHEAD>();
        else if (rem == 2)     wait_asynccnt_le<2>();
        else if (rem == 1)     wait_asynccnt_le<1>();
        else                   wait_asynccnt_le<0>();

        // ds_load_b128 from the just-landed stage.
        i32x4 q = buf[it & (NS - 1)][tid];

        f32x4 f;
        f.x = ((float)q.x - z) * s;   // bit-exact: cvt, sub, mul (matches ref)
        f.y = ((float)q.y - z) * s;
        f.z = ((float)q.z - z) * s;
        f.w = ((float)q.w - z) * s;

        const int col = it * COLS_PER_IT + tid * VEC;
        if (col < INF) {  // predicate only the store; output never re-read
            __builtin_nontemporal_store(f, (f32x4*)(out + rowbase + col));
        }
    }
}

extern "C" void kernel_launch(void* const* d_in, const int* in_sizes, int n_in,
                              void* d_out, int out_size, void* d_ws, size_t ws_size,
                              hipStream_t stream) {
    (void)in_sizes; (void)n_in; (void)d_ws; (void)ws_size; (void)out_size;
    const int*   w  = (const int*)d_in[0];
    const float* s  = (const float*)d_in[1];
    const float* z  = (const float*)d_in[2];
    float*       o  = (float*)d_out;

    dim3 grid(1, OUTF, 1);   // 4096 blocks x 8 waves: ample HBM-latency cover
    dim3 block(TPB, 1, 1);
    AsymmetricWeightsDequantizer_29274497089862_kernel<<<grid, block, 0, stream>>>(w, s, z, o);
}